// KMeans_8684423873057
// MI455X (gfx1250) — compile-verified
//
#include <hip/hip_runtime.h>
#include <hip/hip_bf16.h>

// ---------------- problem constants (match reference) ----------------
constexpr int BGRAPH = 16;
constexpr int NPER   = 2048;
constexpr int MPER   = 1024;             // FPS_RATIO = 0.5
constexpr int NTOT   = BGRAPH * NPER;    // 32768
constexpr int MTOT   = BGRAPH * MPER;    // 16384
constexpr int MAX_ITER = 10;
constexpr float KTOL  = 1e-3f;

typedef float v2f __attribute__((ext_vector_type(2)));
typedef float v8f __attribute__((ext_vector_type(8)));

// ---------------- helpers ----------------
__device__ __forceinline__ unsigned long long packKey(float d, int idx) {
    // non-negative float bits are order-preserving; ~idx makes ties pick smallest idx
    return ((unsigned long long)__float_as_uint(d) << 32) |
           (unsigned long long)(unsigned)(~idx);
}

// ---------------- init: done flag + centroid batch ids ----------------
__global__ void kmeans_init_kernel(int* __restrict__ outBatch, int* __restrict__ flags) {
    int tid = blockIdx.x * blockDim.x + threadIdx.x;
    if (tid < MTOT) outBatch[tid] = tid >> 10;   // tid / MPER
    if (tid == 0) flags[0] = 0;                  // done = false
}

// ---------------- FPS: one workgroup per graph, fully in LDS ----------------
// cent[].w stores the PRE-NEGATED bias  -0.5*||c||^2  so the assign kernel's
// WMMA C-accumulator loads need no runtime negation.
__global__ __launch_bounds__(1024) void fps_kernel(const float* __restrict__ pos,
                                                   float4* __restrict__ cent) {
    __shared__ float4 pts[NPER];          // 32 KB
    __shared__ float  dist[NPER];         // 8 KB
    __shared__ unsigned long long red[32];
    __shared__ float4 chosen;

    const int g    = blockIdx.x;
    const int tid  = threadIdx.x;
    const int lane = tid & 31;
    const int wave = tid >> 5;
    const float* gp = pos + (size_t)g * NPER * 3;

    for (int i = tid; i < NPER; i += 1024)
        pts[i] = make_float4(gp[3*i+0], gp[3*i+1], gp[3*i+2], 0.f);
    __syncthreads();

    float4 p0 = pts[0];
    for (int i = tid; i < NPER; i += 1024) {
        float dx = pts[i].x - p0.x, dy = pts[i].y - p0.y, dz = pts[i].z - p0.z;
        dist[i] = dx*dx + dy*dy + dz*dz;
    }
    if (tid == 0)
        cent[(size_t)g * MPER] =
            make_float4(p0.x, p0.y, p0.z, -0.5f*(p0.x*p0.x + p0.y*p0.y + p0.z*p0.z));
    __syncthreads();

    for (int s = 1; s < MPER; ++s) {
        // per-thread best over its 2 points
        unsigned long long k0 = packKey(dist[tid],        tid);
        unsigned long long k1 = packKey(dist[tid + 1024], tid + 1024);
        unsigned long long k  = k0 > k1 ? k0 : k1;
        #pragma unroll
        for (int m = 16; m > 0; m >>= 1) {
            unsigned long long o = __shfl_xor(k, m, 32);
            k = o > k ? o : k;
        }
        if (lane == 0) red[wave] = k;
        __syncthreads();
        if (wave == 0) {
            unsigned long long kk = red[lane];
            #pragma unroll
            for (int m = 16; m > 0; m >>= 1) {
                unsigned long long o = __shfl_xor(kk, m, 32);
                kk = o > kk ? o : kk;
            }
            if (lane == 0) {
                int idx = (int)(~(unsigned)(kk & 0xffffffffull));
                float4 c = pts[idx];
                chosen = c;
                cent[(size_t)g * MPER + s] =
                    make_float4(c.x, c.y, c.z, -0.5f*(c.x*c.x + c.y*c.y + c.z*c.z));
            }
        }
        __syncthreads();
        float4 c = chosen;
        #pragma unroll
        for (int j = 0; j < 2; ++j) {
            int i = tid + j * 1024;
            float dx = pts[i].x - c.x, dy = pts[i].y - c.y, dz = pts[i].z - c.z;
            float d2 = dx*dx + dy*dy + dz*dz;
            dist[i] = d2 < dist[i] ? d2 : dist[i];
        }
        __syncthreads();
    }
}

// ---------------- zero sums + convergence flag ----------------
__global__ void zero_kernel(float* __restrict__ sums, int* __restrict__ flags) {
    int tid = blockIdx.x * blockDim.x + threadIdx.x;
    if (tid < MTOT * 4) sums[tid] = 0.f;
    if (tid == 0) flags[1] = 0;  // not_conv = 0
}

// ---------------- assignment: WMMA f32 16x16x4, A=centroids B=points ----------------
// score[c,p] = p.c - 0.5*||c||^2  (C preloaded with stored -h); argmax == argmin dist
__global__ __launch_bounds__(256) void assign_kernel(const float* __restrict__ pos,
                                                     const float4* __restrict__ cent,
                                                     float* __restrict__ sums,
                                                     int* __restrict__ outCls,
                                                     const int* __restrict__ flags) {
    if (flags[0]) return;  // frozen after convergence (uniform branch, EXEC stays full)

    __shared__ float4 lc[MPER];          // 16 KB centroid tile cache
    const int g    = blockIdx.x >> 4;
    const int tile = blockIdx.x & 15;
    const int tid  = threadIdx.x;
    const int wave = tid >> 5;
    const int lane = tid & 31;

    for (int i = tid; i < MPER; i += 256) lc[i] = cent[(size_t)g * MPER + i];
    __syncthreads();

    const int prow = lane & 15;               // point column / centroid row within tile
    const int hi   = (lane >> 4) << 3;        // 0 for lanes 0-15 (K={0,1}), 8 for 16-31
    const int p    = tile * 128 + wave * 16 + prow;     // point index within graph
    const float* gp = pos + (size_t)g * NPER * 3;
    const float px = gp[3*p+0], py = gp[3*p+1], pz = gp[3*p+2];

    // B (4x16 f32): lane<16 holds K={0,1} = (x,y); lane>=16 holds K={2,3} = (z,0)
    v2f bmat;
    bmat.x = hi ? pz  : px;
    bmat.y = hi ? 0.f : py;

    float best = -3.402823e38f;
    int   bidx = 0;

    for (int t = 0; t < MPER / 16; ++t) {
        float4 c4 = lc[t * 16 + prow];
        // A (16x4 f32): lane<16 M=prow K={0,1}=(cx,cy); lane>=16 K={2,3}=(cz,0)
        v2f amat;
        amat.x = hi ? c4.z : c4.x;
        amat.y = hi ? 0.f  : c4.y;
        // C = -0.5*||c||^2 (already negated in .w) broadcast across point dimension
        v8f acc;
        #pragma unroll
        for (int r = 0; r < 8; ++r) acc[r] = lc[t * 16 + hi + r].w;

        v8f d = __builtin_amdgcn_wmma_f32_16x16x4_f32(
            /*neg_a=*/false, amat, /*neg_b=*/false, bmat,
            /*c_mod=*/(short)0, acc, /*reuse_a=*/false, /*reuse_b=*/false);

        #pragma unroll
        for (int r = 0; r < 8; ++r) {
            float s = d[r];
            int ci = t * 16 + hi + r;            // ascending -> strict > keeps lowest idx
            if (s > best) { best = s; bidx = ci; }
        }
    }

    // lanes L and L+16 own the same point (rows r and r+8): combine, ties -> lower idx
    float ob = __shfl_xor(best, 16, 32);
    int   oi = __shfl_xor(bidx, 16, 32);
    if (ob > best || (ob == best && oi < bidx)) { best = ob; bidx = oi; }

    if (lane < 16) {
        int gcls = g * MPER + bidx;
        outCls[g * NPER + p] = gcls;
        atomicAdd(&sums[gcls * 4 + 0], px);
        atomicAdd(&sums[gcls * 4 + 1], py);
        atomicAdd(&sums[gcls * 4 + 2], pz);
        atomicAdd(&sums[gcls * 4 + 3], 1.0f);
    }
}

// ---------------- centroid update + convergence test ----------------
__global__ void finalize_kernel(float4* __restrict__ cent,
                                const float* __restrict__ sums,
                                int* __restrict__ flags) {
    int m = blockIdx.x * blockDim.x + threadIdx.x;
    if (m >= MTOT) return;
    float4 c  = cent[m];
    float cnt = sums[m * 4 + 3];
    float nx = c.x, ny = c.y, nz = c.z;
    if (cnt > 0.f) {
        float inv = 1.f / cnt;
        nx = sums[m * 4 + 0] * inv;
        ny = sums[m * 4 + 1] * inv;
        nz = sums[m * 4 + 2] * inv;
    }
    float dx = nx - c.x, dy = ny - c.y, dz = nz - c.z;
    if (dx*dx + dy*dy + dz*dz >= KTOL * KTOL) flags[1] = 1;   // not converged
    cent[m] = make_float4(nx, ny, nz, -0.5f*(nx*nx + ny*ny + nz*nz));
}

__global__ void flag_kernel(int* __restrict__ flags) {
    if (flags[1] == 0) flags[0] = 1;   // done |= converged
}

// ---------------- emit centroid output ----------------
__global__ void copy_cent_kernel(const float4* __restrict__ cent,
                                 float* __restrict__ outCent) {
    int m = blockIdx.x * blockDim.x + threadIdx.x;
    if (m >= MTOT) return;
    float4 c = cent[m];
    outCent[3*m+0] = c.x;
    outCent[3*m+1] = c.y;
    outCent[3*m+2] = c.z;
}

// ---------------- launch ----------------
extern "C" void kernel_launch(void* const* d_in, const int* in_sizes, int n_in,
                              void* d_out, int out_size, void* d_ws, size_t ws_size,
                              hipStream_t stream) {
    (void)in_sizes; (void)n_in; (void)out_size; (void)ws_size;
    const float* pos = (const float*)d_in[0];   // [NTOT,3] f32; d_in[1] (batch) implied

    // outputs concatenated flat: cls int32[NTOT] | cent f32[MTOT*3] | batch int32[MTOT]
    int*   outCls   = (int*)d_out;
    float* outCent  = (float*)d_out + NTOT;
    int*   outBatch = (int*)d_out + NTOT + 3 * MTOT;

    char* ws = (char*)d_ws;
    float4* cent = (float4*)ws;                               // MTOT * 16 B
    float*  sums = (float*)(ws + (size_t)MTOT * 16);          // MTOT * 4 f32
    int*    flags = (int*)(ws + (size_t)MTOT * 32);           // [done, not_conv]

    kmeans_init_kernel<<<(MTOT + 255) / 256, 256, 0, stream>>>(outBatch, flags);
    fps_kernel<<<BGRAPH, 1024, 0, stream>>>(pos, cent);

    for (int it = 0; it < MAX_ITER; ++it) {
        zero_kernel<<<(MTOT * 4 + 255) / 256, 256, 0, stream>>>(sums, flags);
        assign_kernel<<<BGRAPH * 16, 256, 0, stream>>>(pos, cent, sums, outCls, flags);
        finalize_kernel<<<(MTOT + 255) / 256, 256, 0, stream>>>(cent, sums, flags);
        flag_kernel<<<1, 1, 0, stream>>>(flags);
    }
    copy_cent_kernel<<<(MTOT + 255) / 256, 256, 0, stream>>>(cent, outCent);
}